// SAGEReLU_53197464928902
// MI455X (gfx1250) — compile-verified
//
#include <hip/hip_runtime.h>
#include <hip/hip_bf16.h>

typedef __attribute__((ext_vector_type(2))) float v2f;
typedef __attribute__((ext_vector_type(8))) float v8f;

#define NN   100000
#define EE   1600000
#define DIN  128
#define HH   64
#define GG   512

// ---------------------------------------------------------------------------
// Utility: zero a float buffer (grid-stride, capture-safe)
// ---------------------------------------------------------------------------
__global__ void zero_f32(float* __restrict__ p, int n) {
    for (int i = blockIdx.x * blockDim.x + threadIdx.x; i < n; i += gridDim.x * blockDim.x)
        p[i] = 0.0f;
}

// ---------------------------------------------------------------------------
// Degree: count in-edges per dst, then invert with clip(deg,1)
// ---------------------------------------------------------------------------
__global__ void deg_count(const int* __restrict__ ei, float* __restrict__ deg) {
    int e = blockIdx.x * blockDim.x + threadIdx.x;
    if (e < EE) atomicAdd(&deg[ei[EE + e]], 1.0f);
}

__global__ void deg_invert(float* __restrict__ deg) {
    int i = blockIdx.x * blockDim.x + threadIdx.x;
    if (i < NN) deg[i] = 1.0f / fmaxf(deg[i], 1.0f);
}

// ---------------------------------------------------------------------------
// h = relu(x @ W0 + b0)   [N,128]x[128,64] via V_WMMA_F32_16X16X4_F32
// block = 128 threads = 4 waves; wave w owns column tile w; block owns 16 rows
// ---------------------------------------------------------------------------
__global__ void gemm_relu_in(const float* __restrict__ x, const float* __restrict__ W0,
                             const float* __restrict__ b0, float* __restrict__ h) {
    const int row0 = blockIdx.x * 16;
    const int wave = threadIdx.x >> 5;
    const int lane = threadIdx.x & 31;
    const int col0 = wave * 16;
    const int m    = lane & 15;        // A-row / B-col / C-col index for this lane
    const int half = lane >> 4;        // 0: K=0,1   1: K=2,3
    const int ksub = half * 2;

    v8f c = {};
    const float* __restrict__ xrow = x + (size_t)(row0 + m) * DIN;
    for (int k = 0; k < DIN; k += 4) {
        v2f a, b;
        a.x = xrow[k + ksub];
        a.y = xrow[k + ksub + 1];
        b.x = W0[(k + ksub) * HH + col0 + m];
        b.y = W0[(k + ksub + 1) * HH + col0 + m];
        c = __builtin_amdgcn_wmma_f32_16x16x4_f32(false, a, false, b, (short)0, c, false, false);
    }
    const int   n    = col0 + m;
    const float bias = b0[n];
#pragma unroll
    for (int r = 0; r < 8; ++r) {
        int row = row0 + r + half * 8;
        h[(size_t)row * HH + n] = fmaxf(c[r] + bias, 0.0f);
    }
}

// ---------------------------------------------------------------------------
// Edge scatter: agg[dst,:] += h[src,:]   (one thread per (edge,col))
// ---------------------------------------------------------------------------
__global__ void scatter_edges(const int* __restrict__ ei, const float* __restrict__ h,
                              float* __restrict__ agg) {
    int idx = blockIdx.x * blockDim.x + threadIdx.x;
    if (idx >= EE * HH) return;
    int e = idx >> 6;          // HH == 64
    int col = idx & 63;
    int s = ei[e];
    int d = ei[EE + e];
    atomicAdd(&agg[(size_t)d * HH + col], h[(size_t)s * HH + col]);
}

// ---------------------------------------------------------------------------
// h_out = relu((agg*deginv) @ Wl + bl + h_in @ Wr)
// Both GEMMs accumulate into the same WMMA accumulator (fused).
// ---------------------------------------------------------------------------
__global__ void sage_combine(const float* __restrict__ agg, const float* __restrict__ deginv,
                             const float* __restrict__ hin, const float* __restrict__ Wl,
                             const float* __restrict__ bl,  const float* __restrict__ Wr,
                             float* __restrict__ hout) {
    const int row0 = blockIdx.x * 16;
    const int wave = threadIdx.x >> 5;
    const int lane = threadIdx.x & 31;
    const int col0 = wave * 16;
    const int m    = lane & 15;
    const int half = lane >> 4;
    const int ksub = half * 2;

    v8f c = {};
    const float dinv = deginv[row0 + m];
    const float* __restrict__ arow = agg + (size_t)(row0 + m) * HH;
    const float* __restrict__ hrow = hin + (size_t)(row0 + m) * HH;

    // C += (agg * dinv) @ Wl
    for (int k = 0; k < HH; k += 4) {
        v2f a, b;
        a.x = arow[k + ksub] * dinv;
        a.y = arow[k + ksub + 1] * dinv;
        b.x = Wl[(k + ksub) * HH + col0 + m];
        b.y = Wl[(k + ksub + 1) * HH + col0 + m];
        c = __builtin_amdgcn_wmma_f32_16x16x4_f32(false, a, false, b, (short)0, c, false, false);
    }
    // C += h @ Wr
    for (int k = 0; k < HH; k += 4) {
        v2f a, b;
        a.x = hrow[k + ksub];
        a.y = hrow[k + ksub + 1];
        b.x = Wr[(k + ksub) * HH + col0 + m];
        b.y = Wr[(k + ksub + 1) * HH + col0 + m];
        c = __builtin_amdgcn_wmma_f32_16x16x4_f32(false, a, false, b, (short)0, c, false, false);
    }
    const int   n    = col0 + m;
    const float bias = bl[n];
#pragma unroll
    for (int r = 0; r < 8; ++r) {
        int row = row0 + r + half * 8;
        hout[(size_t)row * HH + n] = fmaxf(c[r] + bias, 0.0f);
    }
}

// ---------------------------------------------------------------------------
// Pooling: per-graph counts and feature sums
// ---------------------------------------------------------------------------
__global__ void pool_cnt(const int* __restrict__ batch, float* __restrict__ cnt) {
    int n = blockIdx.x * blockDim.x + threadIdx.x;
    if (n < NN) atomicAdd(&cnt[batch[n]], 1.0f);
}

__global__ void pool_sum(const int* __restrict__ batch, const float* __restrict__ h,
                         float* __restrict__ pooled) {
    int idx = blockIdx.x * blockDim.x + threadIdx.x;
    if (idx >= NN * HH) return;
    int n = idx >> 6;
    int col = idx & 63;
    atomicAdd(&pooled[(size_t)batch[n] * HH + col], h[(size_t)n * HH + col]);
}

// ---------------------------------------------------------------------------
// out = (pooled/cnt) @ W1 + b1   [512,64]x[64,64] via WMMA f32
// ---------------------------------------------------------------------------
__global__ void pool_gemm(const float* __restrict__ pooled, const float* __restrict__ cnt,
                          const float* __restrict__ W1, const float* __restrict__ b1,
                          float* __restrict__ out) {
    const int row0 = blockIdx.x * 16;
    const int wave = threadIdx.x >> 5;
    const int lane = threadIdx.x & 31;
    const int col0 = wave * 16;
    const int m    = lane & 15;
    const int half = lane >> 4;
    const int ksub = half * 2;

    v8f c = {};
    const float cinv = 1.0f / fmaxf(cnt[row0 + m], 1.0f);
    const float* __restrict__ prow = pooled + (size_t)(row0 + m) * HH;
    for (int k = 0; k < HH; k += 4) {
        v2f a, b;
        a.x = prow[k + ksub] * cinv;
        a.y = prow[k + ksub + 1] * cinv;
        b.x = W1[(k + ksub) * HH + col0 + m];
        b.y = W1[(k + ksub + 1) * HH + col0 + m];
        c = __builtin_amdgcn_wmma_f32_16x16x4_f32(false, a, false, b, (short)0, c, false, false);
    }
    const int   n    = col0 + m;
    const float bias = b1[n];
#pragma unroll
    for (int r = 0; r < 8; ++r) {
        int row = row0 + r + half * 8;
        out[(size_t)row * HH + n] = c[r] + bias;
    }
}

// ---------------------------------------------------------------------------
// Host-side orchestration
// ---------------------------------------------------------------------------
extern "C" void kernel_launch(void* const* d_in, const int* in_sizes, int n_in,
                              void* d_out, int out_size, void* d_ws, size_t ws_size,
                              hipStream_t stream) {
    (void)in_sizes; (void)n_in; (void)out_size; (void)ws_size;

    const float* x     = (const float*)d_in[0];
    const int*   ei    = (const int*)  d_in[1];   // [2,E]
    /* d_in[2] = edge_attr, unused by reference */
    const int*   batch = (const int*)  d_in[3];
    const float* W0    = (const float*)d_in[4];
    const float* b0    = (const float*)d_in[5];
    const float* Wl    = (const float*)d_in[6];   // [L,H,H]
    const float* bl    = (const float*)d_in[7];   // [L,H]
    const float* Wr    = (const float*)d_in[8];   // [L,H,H]
    const float* W1    = (const float*)d_in[9];
    const float* b1    = (const float*)d_in[10];
    float*       out   = (float*)d_out;

    // workspace carve-up (floats)
    float* hA     = (float*)d_ws;             // N*H
    float* hB     = hA + (size_t)NN * HH;     // N*H
    float* agg    = hB + (size_t)NN * HH;     // N*H
    float* deg    = agg + (size_t)NN * HH;    // N
    float* pooled = deg + NN;                 // G*H
    float* cnt    = pooled + (size_t)GG * HH; // G

    const int TB = 256;

    // degrees (computed once; same for every layer)
    zero_f32<<<1024, TB, 0, stream>>>(deg, NN);
    deg_count<<<(EE + TB - 1) / TB, TB, 0, stream>>>(ei, deg);
    deg_invert<<<(NN + TB - 1) / TB, TB, 0, stream>>>(deg);

    // h = relu(x @ W0 + b0)
    gemm_relu_in<<<NN / 16, 128, 0, stream>>>(x, W0, b0, hA);

    // 3 SAGE layers, double-buffered
    float* hcur = hA;
    float* hnxt = hB;
    for (int l = 0; l < 3; ++l) {
        zero_f32<<<2048, TB, 0, stream>>>(agg, NN * HH);
        scatter_edges<<<(EE * HH + TB - 1) / TB, TB, 0, stream>>>(ei, hcur, agg);
        sage_combine<<<NN / 16, 128, 0, stream>>>(agg, deg, hcur,
                                                  Wl + (size_t)l * HH * HH,
                                                  bl + (size_t)l * HH,
                                                  Wr + (size_t)l * HH * HH,
                                                  hnxt);
        float* t = hcur; hcur = hnxt; hnxt = t;
    }

    // global mean pool + final linear
    zero_f32<<<(GG * HH + TB - 1) / TB, TB, 0, stream>>>(pooled, GG * HH);
    zero_f32<<<(GG + TB - 1) / TB, TB, 0, stream>>>(cnt, GG);
    pool_cnt<<<(NN + TB - 1) / TB, TB, 0, stream>>>(batch, cnt);
    pool_sum<<<(NN * HH + TB - 1) / TB, TB, 0, stream>>>(batch, hcur, pooled);
    pool_gemm<<<GG / 16, 128, 0, stream>>>(pooled, cnt, W1, b1, out);
}